// IWT_Layer_14448269984448
// MI455X (gfx1250) — compile-verified
//
#include <hip/hip_runtime.h>

// IDWT (8-tap reconstruction, pywt idwt2 core) as WMMA f32 16x16x4 GEMMs.
// x: [16,12,515,515] f32 (channels = C*4 subbands A,H,V,D), out: [16,3,1024,1024] f32.
//
// Polyphase: out[y,x] = sum_{d,e in 0..3} wy[d]*wx[e]*band[(y>>1)+d, (x>>1)+e]
//   even phase taps: lo[6],lo[4],lo[2],lo[0]; odd: lo[7],lo[5],lo[3],lo[1]
//   hi[k] = (-1)^(k+1) lo[k]  ->  even-phase hi = -even-phase lo; odd-phase hi = +odd-phase lo
// Per 16x16 output tile (y0=16*by, x0=16*bx), coeff region is rows 8*by..+10, cols 8*bx..+10.
//   rowsA(16x16) = M_lo(16x12) x cA(12x16) + M_hi x cH      (K padded to 12; col 11 weight = 0)
//   rowsD(16x16) = M_lo x cV + M_hi x cD
//   out(16x16)   = rowsA(16x16) x N_lo(16x16) + rowsD x N_hi (N rows 11..15 weight = 0)
// All boundary taps are in-range by construction (output len = 2n-6); only the
// zero-weight pad rows/cols can exceed 514 -> clamp (finite garbage * 0 = 0).
//
// Each wave owns a 16 x 128 output strip (8 tiles): constant WMMA operands, row
// offsets and the per-lane output pointer are built once and amortized.

typedef float v2f __attribute__((ext_vector_type(2)));
typedef float v8f __attribute__((ext_vector_type(8)));

// Polyphase weight: element [pos-dim p][coeff-dim r] of M (row stage, p=y) or N (col stage, p=x).
__device__ __forceinline__ float mweight(int r, int p, int hif)
{
    int d = r - (p >> 1);
    bool ok = (unsigned)d < 4u;
    float w;
    if (p & 1) {
        // odd phase: lo[7],lo[5],lo[3],lo[1]; hi == +lo here
        w = (d == 0) ? -0.010597401784997278f
          : (d == 1) ?  0.030841381835986965f
          : (d == 2) ? -0.02798376941698385f
          :             0.7148465705525415f;
    } else {
        // even phase: lo[6],lo[4],lo[2],lo[0]; hi == -lo here
        w = (d == 0) ?  0.032883011666982945f
          : (d == 1) ? -0.18703481171888114f
          : (d == 2) ?  0.6308807679295904f
          :             0.23037781330885523f;
        if (hif) w = -w;
    }
    return ok ? w : 0.0f;
}

#define PLANE 265225L   // 515*515
#define WROW  515

__global__ __launch_bounds__(128, 1)
void idwt8_wmma(const float* __restrict__ x, float* __restrict__ out)
{
    const int tid  = threadIdx.x;
    const int lane = tid & 31;
    const int wave = tid >> 5;
    const int l15  = lane & 15;
    const int hi   = lane >> 4;          // which 16-lane half (0/1)

    // one wave = one 16x128 strip (8 tiles); 48 images * 64 rows * 8 strips = 24576 strips
    const int sid = blockIdx.x * 4 + wave;
    const int sx  = sid & 7;             // strip index along x
    const int by  = (sid >> 3) & 63;
    const int img = sid >> 9;            // 0..47 == b*3 + c

    const float* __restrict__ cbase = x + (long)img * 4 * PLANE;      // cA plane; +PLANE per subband
    // per-lane output pointer: all 8 stores per tile become base + constant offset
    float* __restrict__ olane = out + (long)img * (1024L * 1024L)
                                    + (long)(by * 16 + 8 * hi) * 1024 + sx * 128 + l15;

    const int r0 = by * 8;

    // per-lane row offsets (in floats) for the 3 row-stage K-chunks, hoisted
    int rofs0[3], rofs1[3];
    #pragma unroll
    for (int c = 0; c < 3; ++c) {
        int ra = r0 + c * 4 + 2 * hi; if (ra > 514) ra = 514;   // zero-weight pad rows only
        int rb = ra + 1;              if (rb > 514) rb = 514;
        rofs0[c] = ra * WROW;
        rofs1[c] = rb * WROW;
    }

    // ---- constant operands in WMMA register layouts (built once per strip) ---
    // A 16x4 f32 layout: v0 = {K=k0 | K=k0+2 across lane halves}, v1 = {k0+1 | k0+3}, M = lane&15
    // B 4x16 f32 layout assumed symmetric: v0 = {K=k0 | K=k0+2}, v1 = {k0+1 | k0+3}, N = lane&15
    v2f aLo[3], aHi[3];        // row-stage M (16 x 12), 3 K-chunks
    #pragma unroll
    for (int c = 0; c < 3; ++c) {
        int r = c * 4 + 2 * hi;
        aLo[c][0] = mweight(r,     l15, 0);
        aLo[c][1] = mweight(r + 1, l15, 0);
        aHi[c][0] = mweight(r,     l15, 1);
        aHi[c][1] = mweight(r + 1, l15, 1);
    }
    v2f nLo[4], nHi[4];        // col-stage N (16 x 16) as B operand, 4 K-chunks
    #pragma unroll
    for (int c = 0; c < 4; ++c) {
        int r = c * 4 + 2 * hi;
        nLo[c][0] = mweight(r,     l15, 0);
        nLo[c][1] = mweight(r + 1, l15, 0);
        nHi[c][0] = mweight(r,     l15, 1);
        nHi[c][1] = mweight(r + 1, l15, 1);
    }

    // per-wave transpose staging, pitch 17, double-buffered by tile parity
    __shared__ float lbuf[4][2][2][16 * 17];       // [wave][parity][A/D][272]
    const int stA = (8 * hi) * 17 + l15;           // D-layout store base (row v adds v*17)
    const int ldA = l15 * 17 + 2 * hi;             // A-layout load base  (chunk c adds c*4)

    for (int t = 0; t < 8; ++t) {
        const int c0  = sx * 64 + t * 8;
        int col = c0 + l15; if (col > 514) col = 514;   // clamped cols have zero weight downstream

        // ---- row stage: rowsA = M_lo*cA + M_hi*cH ; rowsD = M_lo*cV + M_hi*cD
        v8f rowsA = {0,0,0,0,0,0,0,0};
        v8f rowsD = {0,0,0,0,0,0,0,0};
        #pragma unroll
        for (int c = 0; c < 3; ++c) {
            int o0 = rofs0[c] + col;
            int o1 = rofs1[c] + col;
            v2f bA, bH, bV, bD;
            bA[0] = cbase[o0];             bA[1] = cbase[o1];
            bH[0] = cbase[o0 + PLANE];     bH[1] = cbase[o1 + PLANE];
            bV[0] = cbase[o0 + 2*PLANE];   bV[1] = cbase[o1 + 2*PLANE];
            bD[0] = cbase[o0 + 3*PLANE];   bD[1] = cbase[o1 + 3*PLANE];
            rowsA = __builtin_amdgcn_wmma_f32_16x16x4_f32(false, aLo[c], false, bA, (short)0, rowsA, false, false);
            rowsA = __builtin_amdgcn_wmma_f32_16x16x4_f32(false, aHi[c], false, bH, (short)0, rowsA, false, false);
            rowsD = __builtin_amdgcn_wmma_f32_16x16x4_f32(false, aLo[c], false, bV, (short)0, rowsD, false, false);
            rowsD = __builtin_amdgcn_wmma_f32_16x16x4_f32(false, aHi[c], false, bD, (short)0, rowsD, false, false);
        }

        // ---- transpose C/D layout -> A layout through LDS (parity buffer) ----
        float* bufA = &lbuf[wave][t & 1][0][0];
        float* bufD = &lbuf[wave][t & 1][1][0];
        #pragma unroll
        for (int v = 0; v < 8; ++v) {
            bufA[stA + v * 17] = rowsA[v];
            bufD[stA + v * 17] = rowsD[v];
        }
        __syncthreads();

        // ---- col stage: out = rowsA x N_lo + rowsD x N_hi --------------------
        v8f acc = {0,0,0,0,0,0,0,0};
        #pragma unroll
        for (int c = 0; c < 4; ++c) {
            v2f aA, aD;
            aA[0] = bufA[ldA + c * 4];
            aA[1] = bufA[ldA + c * 4 + 1];
            aD[0] = bufD[ldA + c * 4];
            aD[1] = bufD[ldA + c * 4 + 1];
            acc = __builtin_amdgcn_wmma_f32_16x16x4_f32(false, aA, false, nLo[c], (short)0, acc, false, false);
            acc = __builtin_amdgcn_wmma_f32_16x16x4_f32(false, aD, false, nHi[c], (short)0, acc, false, false);
        }

        // ---- store D (M = v + 8*hi, N = l15); NT: output is write-once -------
        // olane folds img/by/hi/l15; per-tile step is +16 floats; rows are
        // compile-time offsets (v*4096 B) -> single base register + imm offsets
        float* op = olane + t * 16;
        #pragma unroll
        for (int v = 0; v < 8; ++v)
            __builtin_nontemporal_store(acc[v], op + v * 1024);
    }
}

extern "C" void kernel_launch(void* const* d_in, const int* in_sizes, int n_in,
                              void* d_out, int out_size, void* d_ws, size_t ws_size,
                              hipStream_t stream)
{
    const float* xin = (const float*)d_in[0];
    float* o = (float*)d_out;
    // 24576 strips, 4 waves (strips) per 128-thread block
    dim3 grid(6144), block(128);
    hipLaunchKernelGGL(idwt8_wmma, grid, block, 0, stream, xin, o);
}